// GraphConv_2791728742995
// MI455X (gfx1250) — compile-verified
//
#include <hip/hip_runtime.h>

#define N_USERS   50000
#define N_ITEMS   100000
#define N_NODES   150000
#define NNZ       1000000
#define NNZ_HALF  500000
#define D         64
#define N_HOPS    3
#define ND        (N_NODES * D)   /* 9,600,000 */
#define ND_HALF   (ND / 2)        /* 4,800,000 */

#define EB   32    /* edges per batch */
#define BPB  10    /* batches per block: 3125 blocks * 10 * 32 = 1,000,000 */

typedef float v4f __attribute__((ext_vector_type(4)));

struct K2 { unsigned a, b; };

__host__ __device__ constexpr unsigned rotl(unsigned x, int n) {
    return (x << n) | (x >> (32 - n));
}

// Bit-exact JAX/Threefry-2x32 (20 rounds, key injections every 4).
__host__ __device__ constexpr K2 threefry(unsigned k0, unsigned k1, unsigned x0, unsigned x1) {
    unsigned k2 = k0 ^ k1 ^ 0x1BD11BDAu;
    x0 += k0; x1 += k1;
    x0 += x1; x1 = rotl(x1,13); x1 ^= x0;
    x0 += x1; x1 = rotl(x1,15); x1 ^= x0;
    x0 += x1; x1 = rotl(x1,26); x1 ^= x0;
    x0 += x1; x1 = rotl(x1, 6); x1 ^= x0;
    x0 += k1; x1 += k2 + 1u;
    x0 += x1; x1 = rotl(x1,17); x1 ^= x0;
    x0 += x1; x1 = rotl(x1,29); x1 ^= x0;
    x0 += x1; x1 = rotl(x1,16); x1 ^= x0;
    x0 += x1; x1 = rotl(x1,24); x1 ^= x0;
    x0 += k2; x1 += k0 + 2u;
    x0 += x1; x1 = rotl(x1,13); x1 ^= x0;
    x0 += x1; x1 = rotl(x1,15); x1 ^= x0;
    x0 += x1; x1 = rotl(x1,26); x1 ^= x0;
    x0 += x1; x1 = rotl(x1, 6); x1 ^= x0;
    x0 += k0; x1 += k1 + 3u;
    x0 += x1; x1 = rotl(x1,17); x1 ^= x0;
    x0 += x1; x1 = rotl(x1,29); x1 ^= x0;
    x0 += x1; x1 = rotl(x1,16); x1 ^= x0;
    x0 += x1; x1 = rotl(x1,24); x1 ^= x0;
    x0 += k1; x1 += k2 + 4u;
    x0 += x1; x1 = rotl(x1,13); x1 ^= x0;
    x0 += x1; x1 = rotl(x1,15); x1 ^= x0;
    x0 += x1; x1 = rotl(x1,26); x1 ^= x0;
    x0 += x1; x1 = rotl(x1, 6); x1 ^= x0;
    x0 += k2; x1 += k0 + 5u;
    return {x0, x1};
}

// key = key(42); per hop: key, k_e, k_m = split(key, 3)
// split(key,3) = threefry_2x32(key, iota(6)): pairs (0,3),(1,4),(2,5);
// flat=[A.a,B.a,C.a,A.b,B.b,C.b] -> keys (A.a,B.a),(C.a,A.b),(B.b,C.b).
struct HopKeys { unsigned ke0[N_HOPS], ke1[N_HOPS], km0[N_HOPS], km1[N_HOPS]; };

__host__ __device__ constexpr HopKeys make_keys() {
    HopKeys h = {};
    unsigned k0 = 0u, k1 = 42u;
    for (int i = 0; i < N_HOPS; ++i) {
        K2 A = threefry(k0, k1, 0u, 3u);
        K2 B = threefry(k0, k1, 1u, 4u);
        K2 C = threefry(k0, k1, 2u, 5u);
        k0 = A.a;        k1 = B.a;        // new key
        h.ke0[i] = C.a;  h.ke1[i] = A.b;  // k_e (edge dropout)
        h.km0[i] = B.b;  h.km1[i] = C.b;  // k_m (message dropout)
    }
    return h;
}
constexpr HopKeys HK = make_keys();

// JAX uniform: bitcast((bits>>9)|0x3f800000) - 1.0  in [0,1)
__device__ inline float u01(unsigned bits) {
    return __uint_as_float((bits >> 9) | 0x3f800000u) - 1.0f;
}

// ---- hop 0: agg = all_embed; out[:,0,:] = all_embed (b128, NT output) ----
__global__ void k_init(const float* __restrict__ ue, const float* __restrict__ ie,
                       float* __restrict__ agg, float* __restrict__ out) {
    int j = (blockIdx.x * blockDim.x + threadIdx.x) * 4;    // grid covers ND/4 exactly
    int n = j >> 6, d = j & 63;
    const v4f* src = (n < N_USERS) ? (const v4f*)(ue + j)
                                   : (const v4f*)(ie + (j - N_USERS * D));
    v4f v = __builtin_nontemporal_load(src);                // read-once stream
    *(v4f*)(agg + j) = v;                                   // L2-resident working set
    __builtin_nontemporal_store(v, (v4f*)(out + (size_t)n * (4 * D) + d));
}

__global__ void k_zero(v4f* __restrict__ p) {
    p[blockIdx.x * blockDim.x + threadIdx.x] = (v4f){0.f, 0.f, 0.f, 0.f};
}

// ---- SpMM with double-buffered async-to-LDS gather (CDNA5 path) ----------
__global__ __launch_bounds__(256) void k_spmm(const int* __restrict__ rows,
                                              const int* __restrict__ cols,
                                              const float* __restrict__ vals,
                                              const float* __restrict__ aggc,
                                              float* __restrict__ aggn,
                                              unsigned ke0, unsigned ke1) {
    __shared__ float sbuf[2][EB * D];   // 2 x 8KB tile of gathered rows
    __shared__ float sv[2][EB];         // edge weights after edge-dropout
    __shared__ int   srow[2][EB];       // destination rows
    const int t = threadIdx.x;
    const unsigned lbase = (unsigned)(size_t)&sbuf[0][0];   // LDS byte offset of sbuf
    const int batch0 = blockIdx.x * BPB;

    auto issue = [&](int batch, int s) {
        const int ebase = batch * EB;
        #pragma unroll
        for (int k = 0; k < 2; ++k) {
            int idx = t + k * 256;              // 0..511 : (edge, 16B chunk)
            int el = idx >> 4, ch = idx & 15;
            int e = ebase + el;
            const float* g = aggc + (size_t)cols[e] * D + ch * 4;
            unsigned lo = lbase + (unsigned)(((s * (EB * D)) + el * D + ch * 4) * 4);
            asm volatile("global_load_async_to_lds_b128 %0, %1, off"
                         :: "v"(lo), "v"(g) : "memory");
        }
        if (t < EB) {                           // stage v (edge dropout) and row
            int e = ebase + t;
            // prefetch index/value streams two batches ahead (speculative OK at tail)
            __builtin_prefetch(cols + e + 2 * EB, 0, 0);
            __builtin_prefetch(rows + e + 2 * EB, 0, 0);
            __builtin_prefetch(vals + e + 2 * EB, 0, 0);
            float u;
            if (e < NNZ_HALF) { K2 r = threefry(ke0, ke1, (unsigned)e, (unsigned)(e + NNZ_HALF)); u = u01(r.a); }
            else              { K2 r = threefry(ke0, ke1, (unsigned)(e - NNZ_HALF), (unsigned)e); u = u01(r.b); }
            // keep iff floor(0.5 + u) == 1  <=>  u >= 0.5 ; scale 1/(1-0.5)=2
            sv[s][t]   = (u >= 0.5f) ? vals[e] * 2.0f : 0.0f;
            srow[s][t] = rows[e];
        }
    };

    issue(batch0, 0);
    for (int i = 0; i < BPB; ++i) {
        const int s = i & 1;
        if (i + 1 < BPB) {
            issue(batch0 + i + 1, s ^ 1);                     // prefetch next batch
            asm volatile("s_wait_asynccnt 2" ::: "memory");   // batch i retired, i+1 in flight
        } else {
            asm volatile("s_wait_asynccnt 0" ::: "memory");
        }
        __syncthreads();
        #pragma unroll
        for (int k = 0; k < 2; ++k) {
            int idx = t + k * 256;
            int el = idx >> 4, ch = idx & 15;
            float v = sv[s][el];
            if (v != 0.0f) {                     // dropped edges: no atomics at all
                float* dst = aggn + (size_t)srow[s][el] * D + ch * 4;
                const float* m = &sbuf[s][el * D + ch * 4];
                atomicAdd(dst + 0, v * m[0]);
                atomicAdd(dst + 1, v * m[1]);
                atomicAdd(dst + 2, v * m[2]);
                atomicAdd(dst + 3, v * m[3]);
            }
        }
        __syncthreads();                         // protect LDS buffer reuse
    }
}

// ---- message dropout + write hop slice + feed next hop -------------------
// Thread handles 4 counters (i..i+3) -> float4 at i and float4 at i+ND_HALF
// (threefry pairs (c, c+ND_HALF); both output halves consumed).
__global__ void k_drop(const float* __restrict__ aggn, float* __restrict__ aggc,
                       float* __restrict__ out, int hop, unsigned km0, unsigned km1) {
    int i = (blockIdx.x * blockDim.x + threadIdx.x) * 4;    // grid covers ND_HALF/4 exactly
    K2 r0 = threefry(km0, km1, (unsigned)(i + 0), (unsigned)(i + 0 + ND_HALF));
    K2 r1 = threefry(km0, km1, (unsigned)(i + 1), (unsigned)(i + 1 + ND_HALF));
    K2 r2 = threefry(km0, km1, (unsigned)(i + 2), (unsigned)(i + 2 + ND_HALF));
    K2 r3 = threefry(km0, km1, (unsigned)(i + 3), (unsigned)(i + 3 + ND_HALF));
    v4f a0 = __builtin_nontemporal_load((const v4f*)(aggn + i));            // read-once
    v4f a1 = __builtin_nontemporal_load((const v4f*)(aggn + i + ND_HALF));
    v4f v0, v1;
    v0.x = (u01(r0.a) < 0.9f) ? a0.x / 0.9f : 0.0f;   // bernoulli(0.9) keep
    v0.y = (u01(r1.a) < 0.9f) ? a0.y / 0.9f : 0.0f;
    v0.z = (u01(r2.a) < 0.9f) ? a0.z / 0.9f : 0.0f;
    v0.w = (u01(r3.a) < 0.9f) ? a0.w / 0.9f : 0.0f;
    v1.x = (u01(r0.b) < 0.9f) ? a1.x / 0.9f : 0.0f;
    v1.y = (u01(r1.b) < 0.9f) ? a1.y / 0.9f : 0.0f;
    v1.z = (u01(r2.b) < 0.9f) ? a1.z / 0.9f : 0.0f;
    v1.w = (u01(r3.b) < 0.9f) ? a1.w / 0.9f : 0.0f;
    *(v4f*)(aggc + i) = v0;                                 // feeds next hop (keep in L2)
    *(v4f*)(aggc + i + ND_HALF) = v1;
    int n0 = i >> 6, d0 = i & 63;                           // ND_HALF % 64 == 0
    int j = i + ND_HALF, n1 = j >> 6;
    __builtin_nontemporal_store(v0, (v4f*)(out + (size_t)n0 * (4 * D) + (size_t)(hop + 1) * D + d0));
    __builtin_nontemporal_store(v1, (v4f*)(out + (size_t)n1 * (4 * D) + (size_t)(hop + 1) * D + d0));
}

extern "C" void kernel_launch(void* const* d_in, const int* in_sizes, int n_in,
                              void* d_out, int out_size, void* d_ws, size_t ws_size,
                              hipStream_t stream) {
    (void)in_sizes; (void)n_in; (void)out_size; (void)ws_size;
    const float* ue   = (const float*)d_in[0];
    const float* ie   = (const float*)d_in[1];
    const int*   rows = (const int*)d_in[2];
    const int*   cols = (const int*)d_in[3];
    const float* vals = (const float*)d_in[4];
    float* out  = (float*)d_out;
    float* aggA = (float*)d_ws;        // current hop input  [150000 x 64]
    float* aggB = aggA + ND;           // scatter accumulator [150000 x 64]

    k_init<<<(ND / 4) / 256, 256, 0, stream>>>(ue, ie, aggA, out);
    for (int hop = 0; hop < N_HOPS; ++hop) {
        k_zero<<<(ND / 4) / 256, 256, 0, stream>>>((v4f*)aggB);
        k_spmm<<<NNZ / (BPB * EB), 256, 0, stream>>>(rows, cols, vals, aggA, aggB,
                                                     HK.ke0[hop], HK.ke1[hop]);
        k_drop<<<(ND_HALF / 4) / 128, 128, 0, stream>>>(aggB, aggA, out, hop,
                                                        HK.km0[hop], HK.km1[hop]);
    }
}